// ScorePredictor_11905649344604
// MI455X (gfx1250) — compile-verified
//
#include <hip/hip_runtime.h>

// ---------------------------------------------------------------------------
// Edge scorer:  out[e,c] = dot(x[src[e]], W[c,0:128]) + dot(x[dst[e]], W[c,128:256]) + b[c]
//
// Phase 0: build zero-padded B tile (16 cols x 128 K) in workspace.
// Phase 1: per-node projection  s[n][0..3] = [x·W0lo, x·W1lo, x·W0hi, x·W1hi]
//          via V_WMMA_F32_16X16X4_F32 (f32-exact), one 16-node tile per wave.
//          Inner loop = 2x global_load_b64 + v_wmma, no exec-mask games.
// Phase 2: per-edge gather-add from the 1.6 MB (L2-resident) score table.
// ---------------------------------------------------------------------------

typedef __attribute__((ext_vector_type(2))) float v2f;
typedef __attribute__((ext_vector_type(8))) float v8f;

#define D_FEAT 128
#define WAVES_PER_BLOCK 8
#define BPAD_FLOATS (16 * D_FEAT)          // 16 columns x 128 K = 8 KB
#define SCORES_OFFSET_BYTES 32768          // keep scores well clear of Bpad

// Phase 0: Bpad[n*128 + k] = (n<4) ? W[(n&1)*256 + (n>>1)*128 + k] : 0
__global__ __launch_bounds__(256) void build_bpad(
    const float* __restrict__ W, float* __restrict__ bpad)
{
  for (int i = threadIdx.x; i < BPAD_FLOATS; i += 256) {
    const int n = i >> 7;          // column 0..15
    const int k = i & 127;         // K 0..127
    float v = 0.0f;
    if (n < 4) v = W[(size_t)(n & 1) * 256 + (size_t)(n >> 1) * 128 + k];
    bpad[i] = v;
  }
}

__global__ __launch_bounds__(256) void node_proj_wmma(
    const float* __restrict__ x,      // [n_nodes, 128]
    const float* __restrict__ bpad,   // [16, 128] zero-padded B
    float* __restrict__ scores,       // [n_nodes, 4]
    int n_nodes)
{
  const int lane = threadIdx.x & 31;
  const int wave = threadIdx.x >> 5;
  const int tile = blockIdx.x * WAVES_PER_BLOCK + wave;
  if (tile * 16 >= n_nodes) return;            // wave-uniform: EXEC stays all-1s

  // A-matrix (16x4 f32): lane holds row M = lane%16,
  // VGPR0/VGPR1 = K = kbase+0 / kbase+1 with kbase = (lane<16 ? 0 : 2).
  int mrow = tile * 16 + (lane & 15);
  if (mrow >= n_nodes) mrow = n_nodes - 1;     // safe clamp (no-op for 100000 nodes)
  const int koff = (lane < 16) ? 0 : 2;
  const float* xrow = x + (size_t)mrow * D_FEAT + koff;

  // B-matrix (4x16 f32): lane holds column N = lane%16, same K striping as A.
  const int ncol = lane & 15;
  const float* brow = bpad + (size_t)ncol * D_FEAT + koff;

  v8f acc = {};
  #pragma unroll
  for (int kc = 0; kc < D_FEAT / 4; ++kc) {
    v2f a = *(const v2f*)(xrow + kc * 4);      // 8B-aligned
    v2f b = *(const v2f*)(brow + kc * 4);      // unconditional, cache-hot
    // (neg_a, A, neg_b, B, c_mod, C, reuse_a, reuse_b)
    acc = __builtin_amdgcn_wmma_f32_16x16x4_f32(
        false, a, false, b, (short)0, acc, false, false);
  }

  // D layout: VGPR r -> row (r + (lane>=16 ? 8 : 0)) of the tile, column = lane%16.
  // Only columns 0..3 carry real scores.
  if (ncol < 4) {
    const int rowbase = tile * 16 + ((lane >= 16) ? 8 : 0);
    #pragma unroll
    for (int r = 0; r < 8; ++r) {
      const int row = rowbase + r;
      if (row < n_nodes)
        scores[(size_t)row * 4 + ncol] = acc[r];
    }
  }
}

__global__ __launch_bounds__(256) void edge_score(
    const int* __restrict__ src,
    const int* __restrict__ dst,
    const float* __restrict__ scores,  // [n_nodes, 4]
    const float* __restrict__ bias,    // [2]
    float* __restrict__ out,           // [n_edges, 2]
    int n_edges)
{
  const int e = blockIdx.x * blockDim.x + threadIdx.x;
  if (e >= n_edges) return;
  const int s = src[e];
  const int d = dst[e];
  v2f sv = *(const v2f*)(scores + (size_t)s * 4);       // [c0·srcHalf, c1·srcHalf]
  v2f dv = *(const v2f*)(scores + (size_t)d * 4 + 2);   // [c0·dstHalf, c1·dstHalf]
  v2f o;
  o.x = sv.x + dv.x + bias[0];
  o.y = sv.y + dv.y + bias[1];
  *(v2f*)(out + (size_t)e * 2) = o;
}

extern "C" void kernel_launch(void* const* d_in, const int* in_sizes, int n_in,
                              void* d_out, int out_size, void* d_ws, size_t ws_size,
                              hipStream_t stream) {
  const float* x   = (const float*)d_in[0];
  const int*   src = (const int*)d_in[1];
  const int*   dst = (const int*)d_in[2];
  const float* W   = (const float*)d_in[3];
  const float* b   = (const float*)d_in[4];
  float* out    = (float*)d_out;
  float* bpad   = (float*)d_ws;
  float* scores = (float*)((char*)d_ws + SCORES_OFFSET_BYTES); // n_nodes*4*4 B (1.6 MB)

  const int n_nodes = in_sizes[0] / D_FEAT;
  const int n_edges = in_sizes[1];

  build_bpad<<<1, 256, 0, stream>>>(W, bpad);

  const int tiles   = (n_nodes + 15) / 16;
  const int blocks1 = (tiles + WAVES_PER_BLOCK - 1) / WAVES_PER_BLOCK;
  node_proj_wmma<<<blocks1, 256, 0, stream>>>(x, bpad, scores, n_nodes);

  const int blocks2 = (n_edges + 255) / 256;
  edge_score<<<blocks2, 256, 0, stream>>>(src, dst, scores, b, out, n_edges);
}